// BatchSemiHardTripletLoss_9225589752056
// MI455X (gfx1250) — compile-verified
//
#include <hip/hip_runtime.h>
#include <math.h>

// ---------------------------------------------------------------------------
// Batch semi-hard triplet loss, B=8192, D=128.
// Gram = emb @ emb^T via V_WMMA_F32_16X16X4_F32 (exact fp32 matrix path).
// Each wave computes a 2x2 quad of 16x16 tiles -> 1 LDS b64 read per WMMA,
// 4 independent accumulator chains. Distance + masked row reductions fused
// into the GEMM epilogue; two GEMM passes (semi-hard mask needs hardest-pos).
// ---------------------------------------------------------------------------

typedef __attribute__((ext_vector_type(2))) float v2f;
typedef __attribute__((ext_vector_type(8))) float v8f;

#define BATCH   8192
#define DIM     128
#define ROWS    64            // rows per block   (4 row tiles of 16)
#define CHUNK   128           // cols per chunk   (8 col tiles of 16)
#define LSTR    132           // padded LDS row stride: 4-bank step, conflict-free b64
#define MARGIN  0.3f
#define SENTU   0x7F7FFFFFu   // FLT_MAX bit pattern (min-identity, dist >= 0)

// ---------------------------------------------------------------- row norms
__global__ __launch_bounds__(256) void th_row_norms(const float* __restrict__ emb,
                                                    float* __restrict__ sqn) {
  const int row  = blockIdx.x * 8 + (threadIdx.x >> 5);   // one wave per row
  const int lane = threadIdx.x & 31;
  float4 v = ((const float4*)(emb + (size_t)row * DIM))[lane];   // 32 lanes x 4 = 128
  float s = v.x*v.x + v.y*v.y + v.z*v.z + v.w*v.w;
  #pragma unroll
  for (int off = 16; off > 0; off >>= 1) s += __shfl_xor(s, off, 32);
  if (lane == 0) sqn[row] = s;
}

// ---------------------------------------------------------------- main pass
__global__ __launch_bounds__(256) void th_main(const float* __restrict__ emb,
                                               const int*   __restrict__ labels,
                                               const float* __restrict__ sqn,
                                               float* __restrict__ perRow,
                                               float* __restrict__ validf) {
  __shared__ float sA[ROWS  * LSTR];          // 33.8 KB  row strip (resident)
  __shared__ float sB[CHUNK * LSTR];          // 67.6 KB  column chunk (streamed)
  __shared__ unsigned hpS[ROWS], ngS[ROWS], smS[ROWS];

  const int tid   = threadIdx.x;
  const int lane  = tid & 31;
  const int wave  = tid >> 5;                 // 0..7
  const int hi    = lane >> 4;                // 0/1 (half-wave)
  const int l15   = lane & 15;
  const int rtBase = (wave >> 2) * 2;         // row tiles {rtBase, rtBase+1} of 4
  const int ctBase = (wave & 3) * 2;          // col tiles {ctBase, ctBase+1} of 8
  const int rowBase = blockIdx.x * ROWS;

  if (tid < ROWS) { hpS[tid] = 0u; ngS[tid] = SENTU; smS[tid] = SENTU; }

  // stage the block's 64x128 row strip into LDS (coalesced b128 loads)
  for (int idx = tid; idx < ROWS * (DIM / 4); idx += 256) {
    const int r  = idx >> 5;                  // DIM/4 == 32 float4 per row
    const int c4 = idx & 31;
    float4 v = ((const float4*)(emb + (size_t)(rowBase + r) * DIM))[c4];
    float* dst = &sA[r * LSTR + c4 * 4];
    dst[0] = v.x; dst[1] = v.y; dst[2] = v.z; dst[3] = v.w;
  }

  // per-lane row-side constants: lane touches rows M = r + 8*hi of each tile
  float ni[2][8]; int li[2][8]; int ii[2][8];
  #pragma unroll
  for (int rt = 0; rt < 2; ++rt)
    #pragma unroll
    for (int r = 0; r < 8; ++r) {
      const int i = rowBase + (rtBase + rt) * 16 + r + 8 * hi;
      ii[rt][r] = i; ni[rt][r] = sqn[i]; li[rt][r] = labels[i];
    }

  float hpL[2][8], ngL[2][8];
  #pragma unroll
  for (int rt = 0; rt < 2; ++rt)
    #pragma unroll
    for (int r = 0; r < 8; ++r) { hpL[rt][r] = 0.0f; ngL[rt][r] = __uint_as_float(SENTU); }

  // A-frag and B-frag share per-lane addressing (Gram is emb*emb^T):
  // lane holds {x[row][kb], x[row][kb+1]}, row = l15, kb = k0 + 2*hi
  const float* aP0 = &sA[((rtBase + 0) * 16 + l15) * LSTR];
  const float* aP1 = &sA[((rtBase + 1) * 16 + l15) * LSTR];
  const float* bP0 = &sB[((ctBase + 0) * 16 + l15) * LSTR];
  const float* bP1 = &sB[((ctBase + 1) * 16 + l15) * LSTR];

  // ================= PASS 1: hardest positive + overall min negative ======
  for (int cb = 0; cb < BATCH; cb += CHUNK) {
    __syncthreads();                          // prior chunk's sB readers done
    for (int idx = tid; idx < CHUNK * (DIM / 4); idx += 256) {
      const int r  = idx >> 5;
      const int c4 = idx & 31;
      float4 v = ((const float4*)(emb + (size_t)(cb + r) * DIM))[c4];
      float* dst = &sB[r * LSTR + c4 * 4];
      dst[0] = v.x; dst[1] = v.y; dst[2] = v.z; dst[3] = v.w;
    }
    __syncthreads();

    if (cb + CHUNK < BATCH)                   // overlap next chunk's L2->HBM pull
      __builtin_prefetch(emb + (size_t)(cb + CHUNK) * DIM + (size_t)tid * 128, 0, 1);

    v8f acc[2][2] = {{{}, {}}, {{}, {}}};
    #pragma unroll
    for (int k0 = 0; k0 < DIM; k0 += 4) {
      const int kb = k0 + 2 * hi;
      v2f a0 = *(const v2f*)(aP0 + kb);
      v2f a1 = *(const v2f*)(aP1 + kb);
      v2f b0 = *(const v2f*)(bP0 + kb);
      v2f b1 = *(const v2f*)(bP1 + kb);
      acc[0][0] = __builtin_amdgcn_wmma_f32_16x16x4_f32(false, a0, false, b0, (short)0, acc[0][0], false, false);
      acc[0][1] = __builtin_amdgcn_wmma_f32_16x16x4_f32(false, a0, false, b1, (short)0, acc[0][1], false, false);
      acc[1][0] = __builtin_amdgcn_wmma_f32_16x16x4_f32(false, a1, false, b0, (short)0, acc[1][0], false, false);
      acc[1][1] = __builtin_amdgcn_wmma_f32_16x16x4_f32(false, a1, false, b1, (short)0, acc[1][1], false, false);
    }

    #pragma unroll
    for (int ct = 0; ct < 2; ++ct) {
      const int   j  = cb + (ctBase + ct) * 16 + l15;
      const float nj = sqn[j];
      const int   lj = labels[j];
      #pragma unroll
      for (int rt = 0; rt < 2; ++rt)
        #pragma unroll
        for (int r = 0; r < 8; ++r) {
          const float sq = ni[rt][r] + nj - 2.0f * acc[rt][ct][r];
          const float d  = (sq > 0.0f) ? sqrtf(sq) : 0.0f;
          if (li[rt][r] == lj) {
            if (ii[rt][r] != j) hpL[rt][r] = fmaxf(hpL[rt][r], d);  // positive, off-diag
          } else {
            ngL[rt][r] = fminf(ngL[rt][r], d);                      // any negative
          }
        }
    }
  }

  // reduce across the 16 lanes sharing each row, then combine waves in LDS
  #pragma unroll
  for (int rt = 0; rt < 2; ++rt)
    #pragma unroll
    for (int r = 0; r < 8; ++r) {
      float h = hpL[rt][r], n = ngL[rt][r];
      #pragma unroll
      for (int m = 1; m < 16; m <<= 1) {
        h = fmaxf(h, __shfl_xor(h, m, 32));
        n = fminf(n, __shfl_xor(n, m, 32));
      }
      if (l15 == 0) {
        const int rr = (rtBase + rt) * 16 + r + 8 * hi;
        atomicMax(&hpS[rr], __float_as_uint(h));   // bit order == float order (d >= 0)
        atomicMin(&ngS[rr], __float_as_uint(n));
      }
    }
  __syncthreads();

  float hpv[2][8], smL[2][8];
  #pragma unroll
  for (int rt = 0; rt < 2; ++rt)
    #pragma unroll
    for (int r = 0; r < 8; ++r) {
      hpv[rt][r] = __uint_as_float(hpS[(rtBase + rt) * 16 + r + 8 * hi]);
      smL[rt][r] = __uint_as_float(SENTU);
    }

  // ================= PASS 2: semi-hard negative min (needs hardest_pos) ===
  for (int cb = 0; cb < BATCH; cb += CHUNK) {
    __syncthreads();
    for (int idx = tid; idx < CHUNK * (DIM / 4); idx += 256) {
      const int r  = idx >> 5;
      const int c4 = idx & 31;
      float4 v = ((const float4*)(emb + (size_t)(cb + r) * DIM))[c4];
      float* dst = &sB[r * LSTR + c4 * 4];
      dst[0] = v.x; dst[1] = v.y; dst[2] = v.z; dst[3] = v.w;
    }
    __syncthreads();

    if (cb + CHUNK < BATCH)
      __builtin_prefetch(emb + (size_t)(cb + CHUNK) * DIM + (size_t)tid * 128, 0, 1);

    v8f acc[2][2] = {{{}, {}}, {{}, {}}};
    #pragma unroll
    for (int k0 = 0; k0 < DIM; k0 += 4) {
      const int kb = k0 + 2 * hi;
      v2f a0 = *(const v2f*)(aP0 + kb);
      v2f a1 = *(const v2f*)(aP1 + kb);
      v2f b0 = *(const v2f*)(bP0 + kb);
      v2f b1 = *(const v2f*)(bP1 + kb);
      acc[0][0] = __builtin_amdgcn_wmma_f32_16x16x4_f32(false, a0, false, b0, (short)0, acc[0][0], false, false);
      acc[0][1] = __builtin_amdgcn_wmma_f32_16x16x4_f32(false, a0, false, b1, (short)0, acc[0][1], false, false);
      acc[1][0] = __builtin_amdgcn_wmma_f32_16x16x4_f32(false, a1, false, b0, (short)0, acc[1][0], false, false);
      acc[1][1] = __builtin_amdgcn_wmma_f32_16x16x4_f32(false, a1, false, b1, (short)0, acc[1][1], false, false);
    }

    #pragma unroll
    for (int ct = 0; ct < 2; ++ct) {
      const int   j  = cb + (ctBase + ct) * 16 + l15;
      const float nj = sqn[j];
      const int   lj = labels[j];
      #pragma unroll
      for (int rt = 0; rt < 2; ++rt)
        #pragma unroll
        for (int r = 0; r < 8; ++r) {
          const float sq = ni[rt][r] + nj - 2.0f * acc[rt][ct][r];
          const float d  = (sq > 0.0f) ? sqrtf(sq) : 0.0f;
          if (li[rt][r] != lj && d > hpv[rt][r] && d < hpv[rt][r] + MARGIN)
            smL[rt][r] = fminf(smL[rt][r], d);
        }
    }
  }

  #pragma unroll
  for (int rt = 0; rt < 2; ++rt)
    #pragma unroll
    for (int r = 0; r < 8; ++r) {
      float sm = smL[rt][r];
      #pragma unroll
      for (int m = 1; m < 16; m <<= 1) sm = fminf(sm, __shfl_xor(sm, m, 32));
      if (l15 == 0) atomicMin(&smS[(rtBase + rt) * 16 + r + 8 * hi], __float_as_uint(sm));
    }
  __syncthreads();

  // per-row finalize (rows fully owned by this block)
  if (tid < ROWS) {
    const float hp = __uint_as_float(hpS[tid]);
    const float ng = __uint_as_float(ngS[tid]);
    const float sm = __uint_as_float(smS[tid]);
    const bool  hasNeg  = ng < 1e30f;
    const bool  hasSemi = sm < 1e30f;
    const float semiNeg = hasSemi ? sm : ng;
    const bool  valid   = (hp > 0.0f) && hasNeg;
    const float pr      = fmaxf(hp - semiNeg + MARGIN, 0.0f);
    perRow[rowBase + tid] = valid ? pr : 0.0f;
    validf[rowBase + tid] = valid ? 1.0f : 0.0f;
  }
}

// ------------------------------------------------- deterministic final mean
__global__ __launch_bounds__(256) void th_finalize(const float* __restrict__ perRow,
                                                   const float* __restrict__ validf,
                                                   float* __restrict__ out) {
  __shared__ float sT[256], sC[256];
  const int tid = threadIdx.x;
  float t = 0.0f, c = 0.0f;
  for (int i = tid; i < BATCH; i += 256) { t += perRow[i]; c += validf[i]; }
  sT[tid] = t; sC[tid] = c;
  __syncthreads();
  for (int s = 128; s > 0; s >>= 1) {
    if (tid < s) { sT[tid] += sT[tid + s]; sC[tid] += sC[tid + s]; }
    __syncthreads();
  }
  if (tid == 0) out[0] = (sC[0] > 0.0f) ? sT[0] / fmaxf(sC[0], 1.0f) : 0.0f;
}

// ---------------------------------------------------------------------------
extern "C" void kernel_launch(void* const* d_in, const int* in_sizes, int n_in,
                              void* d_out, int out_size, void* d_ws, size_t ws_size,
                              hipStream_t stream) {
  (void)in_sizes; (void)n_in; (void)out_size; (void)ws_size;
  const float* emb    = (const float*)d_in[0];
  const int*   labels = (const int*)d_in[1];
  float* ws     = (float*)d_ws;
  float* sqn    = ws;                 // [0,      8192)
  float* perRow = ws + BATCH;         // [8192,  16384)
  float* validf = ws + 2 * BATCH;     // [16384, 24576)
  float* out    = (float*)d_out;

  th_row_norms<<<BATCH / 8, 256, 0, stream>>>(emb, sqn);
  th_main<<<BATCH / ROWS, 256, 0, stream>>>(emb, labels, sqn, perRow, validf);
  th_finalize<<<1, 256, 0, stream>>>(perRow, validf, out);
}